// Mamba2_43739946943065
// MI455X (gfx1250) — compile-verified
//
#include <hip/hip_runtime.h>
#include <math.h>

typedef float v2f __attribute__((ext_vector_type(2)));
typedef float v8f __attribute__((ext_vector_type(8)));

#define RMS_EPS 1e-5f

__device__ __forceinline__ float silu_f(float x) {
    return x / (1.f + __expf(-x));
}
__device__ __forceinline__ float softplus_f(float x) {
    // numerically stable softplus, matches jax.nn.softplus
    return fmaxf(x, 0.f) + log1pf(__expf(-fabsf(x)));
}

// ---------------------------------------------------------------------------
// RMSNorm over last dim (D=256): one block per row, rows 0..511 = x, 512..1023 = mouth
// Reference divides by sqrt(SUM of squares + eps), then scales by w.
// ---------------------------------------------------------------------------
__global__ void rmsnorm_kernel(const float* __restrict__ x,
                               const float* __restrict__ mouth,
                               const float* __restrict__ w,
                               float* __restrict__ out) {
    __shared__ float red[256];
    const int r = blockIdx.x;      // 0..1023
    const int t = threadIdx.x;     // 0..255
    const float* src = (r < 512) ? (x + (size_t)r * 256)
                                 : (mouth + (size_t)(r - 512) * 256);
    float v = src[t];
    red[t] = v * v;
    __syncthreads();
    for (int s = 128; s > 0; s >>= 1) {
        if (t < s) red[t] += red[t + s];
        __syncthreads();
    }
    float rms = sqrtf(red[0] + RMS_EPS);
    out[(size_t)r * 256 + t] = (v / rms) * w[t];
}

// ---------------------------------------------------------------------------
// Transpose A (512 x 128, d-major) -> AT (128 x 512, n-major) for coalesced SSM reads
// ---------------------------------------------------------------------------
__global__ void transposeA_kernel(const float* __restrict__ A,
                                  float* __restrict__ AT) {
    int i = blockIdx.x * blockDim.x + threadIdx.x;   // over 512*128
    if (i < 512 * 128) {
        int d = i >> 7, n = i & 127;
        AT[n * 512 + d] = A[i];
    }
}

// ---------------------------------------------------------------------------
// fp32 WMMA GEMM:  C[r][n] = act( sum_k X[r][k] * W[n][k] + bias[n] )
//   X row-major [R,K], W row-major [N,K] (i.e. computes X @ W^T), C row-major [R,N]
//   act: 0=none, 1=silu, 2=softplus
// One wave per 16x16 output tile; K stepped by 4 via V_WMMA_F32_16X16X4_F32.
// A-lane layout (ISA 7.12.2): lane L holds A[row0+(L&15)][k0 + 2*(L>>4) + j], j=0,1
// B assumed symmetric: lane L holds B[k0 + 2*(L>>4) + j][col0+(L&15)] = W[col0+(L&15)][...]
// C/D: element v <-> row (v + 8*(L>>4)), col (L&15)
// ---------------------------------------------------------------------------
__global__ void wmma_gemm_kernel(const float* __restrict__ X,
                                 const float* __restrict__ W,
                                 const float* __restrict__ bias,
                                 float* __restrict__ C,
                                 int R, int N, int K, int act) {
    const int wave = (int)((blockIdx.x * blockDim.x + threadIdx.x) >> 5);
    const int lane = threadIdx.x & 31;
    const int ntiles = N >> 4;
    const int total = (R >> 4) * ntiles;
    if (wave >= total) return;               // uniform per wave: EXEC all-1s below

    const int tr = wave / ntiles;
    const int tn = wave - tr * ntiles;
    const int row0 = tr << 4, col0 = tn << 4;
    const int m = lane & 15;
    const int h = lane >> 4;

    const float* arow = X + (size_t)(row0 + m) * K + 2 * h;
    const float* brow = W + (size_t)(col0 + m) * K + 2 * h;

    v8f c = {};
    for (int k0 = 0; k0 < K; k0 += 4) {
        v2f a = *(const v2f*)(arow + k0);
        v2f b = *(const v2f*)(brow + k0);
        c = __builtin_amdgcn_wmma_f32_16x16x4_f32(
                /*neg_a=*/false, a, /*neg_b=*/false, b,
                /*c_mod=*/(short)0, c, /*reuse_a=*/false, /*reuse_b=*/false);
    }

    const float bc = bias[col0 + m];
#pragma unroll
    for (int v = 0; v < 8; ++v) {
        float val = c[v] + bc;
        if (act == 1)      val = silu_f(val);
        else if (act == 2) val = softplus_f(val);
        C[(size_t)(row0 + v + 8 * h) * N + (col0 + m)] = val;
    }
}

// ---------------------------------------------------------------------------
// 3x3 SAME conv over (img, C=2 in-ch, H=256, W=512) + bias + SiLU.
// Row layout of xp/xc: row = img*512 + ch*256 + h, col = w.
// blockIdx.x = img*512 + o*256 + h; 256 threads cover w in two strides.
// ---------------------------------------------------------------------------
__global__ void conv_silu_kernel(const float* __restrict__ xp,
                                 const float* __restrict__ cw,   // (2,2,3,3) OIHW
                                 const float* __restrict__ cb,   // (2,)
                                 float* __restrict__ xc) {
    const int b = blockIdx.x;             // 0..1023
    const int img = b >> 9;
    const int o = (b >> 8) & 1;
    const int hh = b & 255;
    const int t = threadIdx.x;

    float wreg[18];
#pragma unroll
    for (int i = 0; i < 18; ++i) wreg[i] = cw[o * 18 + i];
    const float bo = cb[o];

    for (int w = t; w < 512; w += 256) {
        float acc = bo;
#pragma unroll
        for (int ci = 0; ci < 2; ++ci) {
            const float* base = xp + ((size_t)img * 512 + (size_t)ci * 256) * 512;
#pragma unroll
            for (int kh = 0; kh < 3; ++kh) {
                int hs = hh + kh - 1;
                if (hs < 0 || hs > 255) continue;
#pragma unroll
                for (int kw = 0; kw < 3; ++kw) {
                    int ws = w + kw - 1;
                    if (ws < 0 || ws > 511) continue;
                    acc = fmaf(base[(size_t)hs * 512 + ws], wreg[ci * 9 + kh * 3 + kw], acc);
                }
            }
        }
        xc[(size_t)b * 512 + w] = silu_f(acc);
    }
}

// ---------------------------------------------------------------------------
// Fused SSM epilogue + SiLU.
//   out[r][d] = silu( sum_n Bm[r][n]*exp(delta[r][d]*A[d][n])
//                     + xc[r][d]*delta[r][d]*sum_n Bm[r][n]^2 )
// One block per row r; Bm row staged in LDS; AT[n][d] coalesced over d.
// ---------------------------------------------------------------------------
__global__ void ssm_kernel(const float* __restrict__ xc,
                           const float* __restrict__ delta,
                           const float* __restrict__ Bm,    // [1024,128]
                           const float* __restrict__ AT,    // [128,512]
                           float* __restrict__ out) {
    __shared__ float bm[128];
    const int r = blockIdx.x;
    const int t = threadIdx.x;
    if (t < 128) bm[t] = Bm[(size_t)r * 128 + t];
    __syncthreads();

    for (int d = t; d < 512; d += 256) {
        const float dl = delta[(size_t)r * 512 + d];
        const float xv = xc[(size_t)r * 512 + d];
        float acc = 0.f, sb2 = 0.f;
#pragma unroll 4
        for (int n = 0; n < 128; ++n) {
            const float bv = bm[n];
            acc = fmaf(bv, __expf(dl * AT[n * 512 + d]), acc);
            sb2 = fmaf(bv, bv, sb2);
        }
        out[(size_t)r * 512 + d] = silu_f(acc + xv * dl * sb2);
    }
}

// ---------------------------------------------------------------------------
// comb[r][k] = (ssm_x[r][k] + ssm_m[r][k]) * res[r][k]   (r in [0,512), k in [0,512))
// ssm rows 0..511 = x image, rows 512..1023 = mouth image.
// ---------------------------------------------------------------------------
__global__ void comb_kernel(const float* __restrict__ s,
                            const float* __restrict__ res,
                            float* __restrict__ comb) {
    int i = blockIdx.x * blockDim.x + threadIdx.x;   // over 512*512
    if (i < 512 * 512)
        comb[i] = (s[i] + s[512 * 512 + i]) * res[i];
}

// ---------------------------------------------------------------------------

extern "C" void kernel_launch(void* const* d_in, const int* in_sizes, int n_in,
                              void* d_out, int out_size, void* d_ws, size_t ws_size,
                              hipStream_t stream) {
    const float* x      = (const float*)d_in[0];
    const float* mouth  = (const float*)d_in[1];
    const float* norm_w = (const float*)d_in[2];
    const float* inp_W  = (const float*)d_in[3];
    const float* inp_b  = (const float*)d_in[4];
    const float* out_W  = (const float*)d_in[5];
    const float* out_b  = (const float*)d_in[6];
    const float* Dlin_W = (const float*)d_in[7];
    const float* Dlin_b = (const float*)d_in[8];
    const float* conv_W = (const float*)d_in[9];
    const float* conv_b = (const float*)d_in[10];
    const float* fc1_W  = (const float*)d_in[11];
    const float* fc1_b  = (const float*)d_in[12];
    const float* fc2_W  = (const float*)d_in[13];
    const float* fc2_b  = (const float*)d_in[14];
    const float* A      = (const float*)d_in[15];
    float* out = (float*)d_out;

    // workspace carve-up (floats)
    float* ws    = (float*)d_ws;
    float* xn    = ws;                     // [1024,256] rmsnorm(x)||rmsnorm(mouth)
    float* xp    = xn    + 1024 * 256;     // [1024,512] inp projection
    float* xc    = xp    + 1024 * 512;     // [1024,512] conv+silu
    float* res   = xc    + 1024 * 512;     // [ 512,512] silu(Dlin(xn))
    float* Bmbuf = res   +  512 * 512;     // [1024,128] fc2
    float* dlt   = Bmbuf + 1024 * 128;     // [1024,512] softplus(fc1)
    float* ssm   = dlt   + 1024 * 512;     // [1024,512] ssm+silu
    float* comb  = ssm   + 1024 * 512;     // [ 512,512]
    float* AT    = comb  +  512 * 512;     // [ 128,512]

    auto gemm = [&](const float* X, const float* W, const float* b, float* C,
                    int R, int N, int K, int act) {
        int tiles  = (R >> 4) * (N >> 4);     // one wave per tile
        int blocks = (tiles + 3) / 4;         // 4 waves (128 threads) per block
        wmma_gemm_kernel<<<blocks, 128, 0, stream>>>(X, W, b, C, R, N, K, act);
    };

    rmsnorm_kernel<<<1024, 256, 0, stream>>>(x, mouth, norm_w, xn);
    transposeA_kernel<<<(512 * 128 + 255) / 256, 256, 0, stream>>>(A, AT);

    gemm(xn, inp_W, inp_b, xp, 1024, 512, 256, /*act=*/0);        // input proj (x & mouth)
    gemm(xn, Dlin_W, Dlin_b, res, 512, 512, 256, /*act=*/1);      // res = silu(Dlin(x_n))

    conv_silu_kernel<<<1024, 256, 0, stream>>>(xp, conv_W, conv_b, xc);

    gemm(xc, fc2_W, fc2_b, Bmbuf, 1024, 128, 512, /*act=*/0);     // Bm = fc2(xc)
    gemm(xc, fc1_W, fc1_b, dlt,   1024, 512, 512, /*act=*/2);     // delta = softplus(fc1(xc))

    ssm_kernel<<<1024, 256, 0, stream>>>(xc, dlt, Bmbuf, AT, ssm);
    comb_kernel<<<(512 * 512 + 255) / 256, 256, 0, stream>>>(ssm, res, comb);

    gemm(comb, out_W, out_b, out, 512, 256, 512, /*act=*/0);      // final projection
}